// LlamaAttention_82995948028400
// MI455X (gfx1250) — compile-verified
//
#include <hip/hip_runtime.h>
#include <math.h>

// Decode attention (B=8,H=32,S=4096,D=128, fp32), flash-decoding style.
// BW-bound: ~1.07 GB of K/V traffic -> ~46us floor at 23.3 TB/s.
// QK^T uses v_wmma_f32_16x16x4_f32; K tiles are staged to LDS with the CDNA5
// async DMA path (global_load_async_to_lds_b128 + s_wait_asynccnt), double
// buffered per wave, so global K traffic is perfectly coalesced 512B rows.

typedef float v2f __attribute__((ext_vector_type(2)));
typedef float v8f __attribute__((ext_vector_type(8)));
typedef int v4i __attribute__((vector_size(4 * sizeof(int))));

#define B_ 8
#define S_ 4096
#define H_ 32
#define D_ 128
#define NSPLIT 4
#define WAVES 8
#define BLOCK (WAVES * 32)
#define SPLIT_LEN (S_ / NSPLIT)          // 1024
#define TILES_PER_SPLIT (SPLIT_LEN / 16) // 64
#define WS_STRIDE (D_ + 2)               // m, l, out[128] per (bh, split)
#define KROW 132  // 128 + 4 dword pad -> conflict-free ds_load_b64 fragments

__device__ __forceinline__ void async_cp_b128(const float* g, float* l) {
  // global -> LDS async copy, 16B per lane (512B per wave = one K row)
  __builtin_amdgcn_global_load_async_to_lds_b128(
      (__attribute__((address_space(1))) v4i*)g,
      (__attribute__((address_space(3))) v4i*)l, 0, 0);
}

// One 16-row QK tile from the LDS-staged K tile:
// C = K_tile(16x128) x q(128), q replicated over all 16 B columns.
// A layout (32-bit 16x4): lane l<16 holds A[l][0],A[l][1]; lane l+16 holds
// A[l][2],A[l][3]; dsel = (lane>>4)*2.
template <bool LAST>
__device__ __forceinline__ v8f qk_tile_lds(const float* kt, const float* qr,
                                           const float* kr, int dsel, int lm) {
  v8f c = {};
  const float* a_row = kt + lm * KROW + dsel;
#pragma unroll
  for (int kk = 0; kk < 32; ++kk) {
    const int col = 4 * kk + dsel;
    float2 g = *(const float2*)(a_row + 4 * kk);
    if (LAST) {
      // Row s = S-1 must come from the RoPE'd new k, not the cache.
      float2 r = *(const float2*)(kr + col);
      if (lm == 15) g = r;
    }
    v2f a; a[0] = g.x; a[1] = g.y;
    float2 qq = *(const float2*)(qr + col);
    v2f bq; bq[0] = qq.x; bq[1] = qq.y;
    c = __builtin_amdgcn_wmma_f32_16x16x4_f32(false, a, false, bq,
                                              (short)0, c, false, false);
  }
  return c;
}

__global__ __launch_bounds__(BLOCK) void attn_partial(
    const float* __restrict__ qkv,     // (B,3,H,D)
    const float* __restrict__ kcache,  // (B,1,S,H,D)
    const float* __restrict__ vcache,  // (B,1,S,H,D)
    float* __restrict__ ws) {
  __shared__ __align__(16) float ktile[WAVES][2][16 * KROW];  // 135KB staging
  __shared__ __align__(16) float qr[D_];  // RoPE'd q, pre-scaled by 1/sqrt(D)
  __shared__ __align__(16) float kr[D_];  // RoPE'd new k
  __shared__ float wm[WAVES], wl[WAVES];
  __shared__ __align__(16) float wout[WAVES][D_];

  const int bh = blockIdx.x / NSPLIT;
  const int split = blockIdx.x % NSPLIT;
  const int b = bh / H_, h = bh % H_;
  const int t = threadIdx.x;

  // --- RoPE at position S-1 for q (threads 0..127) and k (threads 128..255)
  {
    const int which = t >> 7;  // 0 -> q, 1 -> k
    const int d = t & (D_ - 1);
    const float expo = (float)(d & ~1) / (float)D_;
    const float inv = __powf(10000.0f, -expo);
    const float ang = (float)(S_ - 1) * inv;
    float sn, cs;
    __sincosf(ang, &sn, &cs);
    const float* src = qkv + ((size_t)(b * 3 + which) * H_ + h) * D_;
    const float x = src[d];
    const float xr = (d & 1) ? src[d - 1] : -src[d + 1];
    const float val = x * cs + xr * sn;
    if (which == 0)
      qr[d] = val * 0.08838834764831845f;  // fold 1/sqrt(128) into q
    else
      kr[d] = val;
  }
  __syncthreads();

  const int wave = t >> 5;
  const int lane = t & 31;
  const int lm = lane & 15;
  const int dsel = (lane >> 4) << 1;

  float m = -3.402823466e38f;
  float lsum = 0.0f;
  float acc[4] = {0.f, 0.f, 0.f, 0.f};  // lane owns d = 4*lane .. 4*lane+3
  const int s_begin = split * SPLIT_LEN;

  float* buf0 = &ktile[wave][0][0];
  float* buf1 = &ktile[wave][1][0];

  auto tile_kbase = [&](int ti2) {
    const int s0 = s_begin + ti2 * 16;
    return kcache + ((size_t)((size_t)b * S_ + s0) * H_ + h) * D_;
  };
  auto prefetch_tile = [&](int ti2, float* buf) {
    const float* base = tile_kbase(ti2);
#pragma unroll
    for (int r = 0; r < 16; ++r)
      async_cp_b128(base + (size_t)r * (H_ * D_) + 4 * lane,
                    buf + r * KROW + 4 * lane);
  };

  prefetch_tile(wave, buf0);  // 16 async b128 ops outstanding
  int parity = 0;

  for (int ti = wave; ti < TILES_PER_SPLIT; ti += WAVES) {
    const int tn = ti + WAVES;
    if (tn < TILES_PER_SPLIT) {
      prefetch_tile(tn, parity ? buf0 : buf1);       // 32 outstanding
      __builtin_amdgcn_s_wait_asynccnt(16);          // current tile landed
    } else {
      __builtin_amdgcn_s_wait_asynccnt(0);
    }
    asm volatile("" ::: "memory");  // keep ds reads below the wait
    const float* cur = parity ? buf1 : buf0;
    parity ^= 1;

    const int s0 = s_begin + ti * 16;
    const bool lastTile = (s0 == S_ - 16);

    v8f c = lastTile ? qk_tile_lds<true>(cur, qr, kr, dsel, lm)
                     : qk_tile_lds<false>(cur, qr, kr, dsel, lm);
    // lane<16: c[j] = score(s0+j); lane>=16: c[j] = score(s0+8+j)

    // --- online softmax (wave-uniform stats)
    float tmax = c[0];
#pragma unroll
    for (int j = 1; j < 8; ++j) tmax = fmaxf(tmax, c[j]);
    tmax = fmaxf(tmax, __shfl_xor(tmax, 16, 32));
    const float mnew = fmaxf(m, tmax);
    const float alpha = __expf(m - mnew);
    float p[8];
    float ts = 0.0f;
#pragma unroll
    for (int j = 0; j < 8; ++j) {
      p[j] = __expf(c[j] - mnew);
      ts += p[j];
    }
    ts += __shfl_xor(ts, 16, 32);
    lsum = lsum * alpha + ts;
    m = mnew;
#pragma unroll
    for (int j = 0; j < 4; ++j) acc[j] *= alpha;

    // --- P.V : lane owns 4 contiguous d's -> 512B coalesced row loads
#pragma unroll
    for (int s = 0; s < 16; ++s) {
      const float ps = __shfl(p[s & 7], (s >> 3) << 4, 32);
      const int sg = s0 + s;
      const float* vrow =
          (lastTile && s == 15)
              ? (qkv + ((size_t)(b * 3 + 2) * H_ + h) * D_)  // new v
              : (vcache + ((size_t)((size_t)b * S_ + sg) * H_ + h) * D_);
      const float4 vv = *(const float4*)(vrow + 4 * lane);
      acc[0] += ps * vv.x;
      acc[1] += ps * vv.y;
      acc[2] += ps * vv.z;
      acc[3] += ps * vv.w;
    }
  }

  // --- combine the 8 waves of this workgroup
  if (lane == 0) { wm[wave] = m; wl[wave] = lsum; }
#pragma unroll
  for (int j = 0; j < 4; ++j) wout[wave][4 * lane + j] = acc[j];
  __syncthreads();

  if (t < D_) {
    const int d = t;
    float mstar = wm[0];
    for (int w = 1; w < WAVES; ++w) mstar = fmaxf(mstar, wm[w]);
    float num = 0.0f, den = 0.0f;
    for (int w = 0; w < WAVES; ++w) {
      const float a = __expf(wm[w] - mstar);
      num += a * wout[w][d];
      den += a * wl[w];
    }
    float* dst = ws + (size_t)(bh * NSPLIT + split) * WS_STRIDE;
    if (d == 0) { dst[0] = mstar; dst[1] = den; }
    dst[2 + d] = num;
  }
}

// Combine NSPLIT partials per (b,h) and normalize.
__global__ __launch_bounds__(D_) void attn_combine(const float* __restrict__ ws,
                                                   float* __restrict__ out) {
  const int bh = blockIdx.x;
  const int d = threadIdx.x;
  float mstar = -3.402823466e38f;
  for (int s = 0; s < NSPLIT; ++s)
    mstar = fmaxf(mstar, ws[(size_t)(bh * NSPLIT + s) * WS_STRIDE]);
  float num = 0.0f, den = 0.0f;
  for (int s = 0; s < NSPLIT; ++s) {
    const float* p = ws + (size_t)(bh * NSPLIT + s) * WS_STRIDE;
    const float a = __expf(p[0] - mstar);
    num += a * p[2 + d];
    den += a * p[1];
  }
  out[(size_t)bh * D_ + d] = num / den;  // (B,H,1,D) flat == bh*D + d
}

extern "C" void kernel_launch(void* const* d_in, const int* in_sizes, int n_in,
                              void* d_out, int out_size, void* d_ws,
                              size_t ws_size, hipStream_t stream) {
  (void)in_sizes; (void)n_in; (void)out_size; (void)ws_size;
  const float* qkv = (const float*)d_in[0];
  const float* kc = (const float*)d_in[1];
  const float* vc = (const float*)d_in[2];
  // d_in[3] = current_seq_len (==S_), d_in[4] = layer_idx (==0): compile-time.
  float* ws = (float*)d_ws;  // needs B*H*NSPLIT*130 floats = 532,480 bytes

  attn_partial<<<B_ * H_ * NSPLIT, BLOCK, 0, stream>>>(qkv, kc, vc, ws);
  attn_combine<<<B_ * H_, D_, 0, stream>>>(ws, (float*)d_out);
}